// SRNO_53403623358971
// MI455X (gfx1250) — compile-verified
//
#include <hip/hip_runtime.h>
#include <math.h>

typedef __attribute__((ext_vector_type(2))) float v2f;
typedef __attribute__((ext_vector_type(8))) float v8f;

// ---------------- problem constants (from reference setup_inputs) ------------
constexpr int  B_    = 2;
constexpr int  HLR   = 96,  WLR = 96;
constexpr int  NF    = 64;
constexpr int  HHI   = 192, WHI = 192;
constexpr int  NPIX  = HHI * WHI;      // 36864 tokens
constexpr int  WID   = 256;
constexpr int  NHEAD = 16;
constexpr int  CIN0  = 272;            // 266 padded up to multiple of 16
constexpr int  PLR   = HLR * WLR;      // 9216
constexpr int  PADW  = 98;
constexpr long PADA  = (long)PADW * PADW; // 9604

__device__ __forceinline__ float gelu_exact(float x) {
    return 0.5f * x * (1.0f + erff(x * 0.70710678118654752440f));
}

__device__ __forceinline__ v8f wmma4(v2f a, v2f b, v8f c) {
    // V_WMMA_F32_16X16X4_F32 : D(16x16,f32) = A(16x4,f32) * B(4x16,f32) + C
    return __builtin_amdgcn_wmma_f32_16x16x4_f32(false, a, false, b, (short)0, c,
                                                 false, false);
}

// ---------------- zero-pad NCHW 96x96 -> 98x98 -------------------------------
__global__ void k_pad(const float* __restrict__ src, float* __restrict__ dst,
                      int C, int total) {
    int idx = blockIdx.x * 256 + threadIdx.x;
    if (idx >= total) return;
    int x  = idx % PADW;
    int y  = (idx / PADW) % PADW;
    int bc = idx / (PADW * PADW);          // b*C + c
    float v = 0.0f;
    if (y >= 1 && y <= HLR && x >= 1 && x <= WLR)
        v = src[((size_t)bc * HLR + (y - 1)) * WLR + (x - 1)];
    dst[idx] = v;
}

// ---------------- head conv 3->64 (direct, tiny) -----------------------------
__global__ void k_head(const float* __restrict__ ipad,   // (B,3,98,98)
                       const float* __restrict__ w,      // (64,3,3,3)
                       const float* __restrict__ bias,
                       float* __restrict__ xh, float* __restrict__ res) {
    int idx = blockIdx.x * 256 + threadIdx.x;
    if (idx >= B_ * NF * PLR) return;
    int x  = idx % WLR;
    int y  = (idx / WLR) % HLR;
    int co = (idx / PLR) % NF;
    int b  =  idx / (PLR * NF);
    float acc = bias[co];
    for (int ci = 0; ci < 3; ++ci) {
        const float* ip = ipad + ((size_t)(b * 3 + ci)) * PADA;
        const float* wp = w + ((size_t)(co * 3 + ci)) * 9;
        #pragma unroll
        for (int ky = 0; ky < 3; ++ky)
            #pragma unroll
            for (int kx = 0; kx < 3; ++kx)
                acc += ip[(y + ky) * PADW + x + kx] * wp[ky * 3 + kx];
    }
    xh[idx]  = acc;
    res[idx] = acc;
}

// ---------------- 3x3 conv 64->64, implicit-GEMM WMMA, tap-major K -----------
// MODE 0: out = relu(conv + bias);  MODE 1: out = conv + bias + add
// Each wave computes a 16co x 32px tile (two 16x16 WMMA tiles sharing A).
template<int MODE>
__global__ void __launch_bounds__(256)
k_conv3(const float* __restrict__ inpad,  // (B,64,98,98)
        const float* __restrict__ w,      // (64,64,3,3)
        const float* __restrict__ bias,
        const float* __restrict__ add,
        float* __restrict__ out) {
    __shared__ float wl[16][577];          // reordered: k' = tap*64 + ci
    const int tid = threadIdx.x;
    const int cb  = blockIdx.y * 16;
    const int b   = blockIdx.z;
    for (int i = tid; i < 16 * 576; i += 256) {
        int m = i / 576, k = i - m * 576;
        int tap = k >> 6, ci = k & 63;
        wl[m][k] = w[(size_t)(cb + m) * 576 + ci * 9 + tap];
    }
    __syncthreads();

    const int wave = tid >> 5, lane = tid & 31;
    const int half = lane >> 4, ln = lane & 15;
    const int pair = blockIdx.x * 8 + wave;      // 0..287 (3 pairs per row)
    const int ty   = pair / 3;
    const int tx   = (pair - ty * 3) * 32;
    const float* ip = inpad + (size_t)b * NF * PADA;

    v8f acc0 = {}, acc1 = {};
    for (int tap = 0; tap < 9; ++tap) {
        const int ky = tap / 3, kx = tap - ky * 3;
        const float* bp   = ip + (size_t)(ty + ky) * PADW + tx + kx + ln;
        const float* wrow = &wl[ln][tap * 64];
        #pragma unroll
        for (int c0 = 0; c0 < 64; c0 += 16) {
            float ax[4], ay[4], b0x[4], b0y[4], b1x[4], b1y[4];
            #pragma unroll
            for (int u = 0; u < 4; ++u) {
                const int kk = c0 + 4 * u + half * 2;
                ax[u]  = wrow[kk];
                ay[u]  = wrow[kk + 1];
                b0x[u] = bp[(size_t)kk * PADA];
                b0y[u] = bp[(size_t)(kk + 1) * PADA];
                b1x[u] = bp[(size_t)kk * PADA + 16];
                b1y[u] = bp[(size_t)(kk + 1) * PADA + 16];
            }
            #pragma unroll
            for (int u = 0; u < 4; ++u) {
                v2f a  = {ax[u],  ay[u]};
                v2f f0 = {b0x[u], b0y[u]};
                v2f f1 = {b1x[u], b1y[u]};
                acc0 = wmma4(a, f0, acc0);
                acc1 = wmma4(a, f1, acc1);
            }
        }
    }
    #pragma unroll
    for (int r = 0; r < 8; ++r) {
        const int c = cb + r + half * 8;
        float v0 = acc0[r] + bias[c];
        float v1 = acc1[r] + bias[c];
        size_t o = ((size_t)b * NF + c) * PLR + (size_t)ty * WLR + tx + ln;
        if (MODE == 0) {
            v0 = fmaxf(v0, 0.0f);
            v1 = fmaxf(v1, 0.0f);
        } else {
            v0 += add[o];
            v1 += add[o + 16];
        }
        out[o]      = v0;
        out[o + 16] = v1;
    }
}

// ---------------- pad conv00 weights 266 -> 272 cols -------------------------
__global__ void k_padw(const float* __restrict__ w, float* __restrict__ wp) {
    int idx = blockIdx.x * 256 + threadIdx.x;
    if (idx >= WID * CIN0) return;
    int k = idx % CIN0, co = idx / CIN0;
    wp[idx] = (k < 266) ? w[(size_t)co * 266 + k] : 0.0f;
}

// ---------------- generic channel-major GEMM (WMMA f32) ----------------------
// Y[b,co,p] = act( sum_ci W[co,ci]*X[b,ci,p] + bias[co] ) (+ add[b,co,p])
// Ci must be a multiple of 16. Each wave: 16co x 32 tokens.
template<int ACT, bool HASADD>
__global__ void __launch_bounds__(256)
k_gemm(const float* __restrict__ W, const float* __restrict__ bias,
       const float* __restrict__ X, const float* __restrict__ add,
       float* __restrict__ Y, int Ci, int Co) {
    __shared__ float wl[16][273];
    const int tid = threadIdx.x;
    const int cb  = blockIdx.y * 16;
    const int b   = blockIdx.z;
    for (int i = tid; i < 16 * Ci; i += 256) {
        int m = i / Ci, k = i - m * Ci;
        wl[m][k] = W[(size_t)(cb + m) * Ci + k];
    }
    __syncthreads();

    const int wave = tid >> 5, lane = tid & 31;
    const int half = lane >> 4, ln = lane & 15;
    const int p0   = (blockIdx.x * 8 + wave) * 32;
    const float* Xp = X + (size_t)b * Ci * NPIX + p0 + ln;

    v8f acc0 = {}, acc1 = {};
    for (int k = 0; k < Ci; k += 16) {
        float ax[4], ay[4], b0x[4], b0y[4], b1x[4], b1y[4];
        #pragma unroll
        for (int u = 0; u < 4; ++u) {
            const int kk = k + 4 * u + half * 2;
            ax[u]  = wl[ln][kk];
            ay[u]  = wl[ln][kk + 1];
            b0x[u] = Xp[(size_t)kk * NPIX];
            b0y[u] = Xp[(size_t)(kk + 1) * NPIX];
            b1x[u] = Xp[(size_t)kk * NPIX + 16];
            b1y[u] = Xp[(size_t)(kk + 1) * NPIX + 16];
        }
        #pragma unroll
        for (int u = 0; u < 4; ++u) {
            v2f a  = {ax[u],  ay[u]};
            v2f f0 = {b0x[u], b0y[u]};
            v2f f1 = {b1x[u], b1y[u]};
            acc0 = wmma4(a, f0, acc0);
            acc1 = wmma4(a, f1, acc1);
        }
    }
    #pragma unroll
    for (int r = 0; r < 8; ++r) {
        const int c = cb + r + half * 8;
        float v0 = acc0[r] + bias[c];
        float v1 = acc1[r] + bias[c];
        if (ACT == 1) { v0 = gelu_exact(v0); v1 = gelu_exact(v1); }
        size_t o = ((size_t)b * Co + c) * NPIX + p0 + ln;
        if (HASADD) { v0 += add[o]; v1 += add[o + 16]; }
        Y[o]      = v0;
        Y[o + 16] = v1;
    }
}

// ---------------- grid-sample + relative-coord feature assembly --------------
// grid: (B, 272, NPIX); ch 0..7 rel coords, 8..263 scaled feats, 264..265 cell,
// 266..271 zero padding.
__global__ void k_make_grid(const float* __restrict__ coord,  // (B,192,192,2)
                            const float* __restrict__ cell,   // (B,2)
                            const float* __restrict__ feat,   // (B,64,96,96)
                            float* __restrict__ grid) {
    int p = blockIdx.x * 256 + threadIdx.x;
    int b = blockIdx.y;
    if (p >= NPIX) return;
    const float cy = coord[((size_t)b * NPIX + p) * 2 + 0];
    const float cx = coord[((size_t)b * NPIX + p) * 2 + 1];
    const float r  = 1.0f / 96.0f, eps = 1e-6f;

    float areas[4]; int iys[4], ixs[4];
    float tot = 0.0f;
    #pragma unroll
    for (int t = 0; t < 4; ++t) {
        const float vx = (t < 2) ? -1.0f : 1.0f;
        const float vy = (t & 1) ? 1.0f : -1.0f;
        float c0 = cy + vx * r + eps;
        float c1 = cx + vy * r + eps;
        c0 = fminf(fmaxf(c0, -1.0f + 1e-6f), 1.0f - 1e-6f);
        c1 = fminf(fmaxf(c1, -1.0f + 1e-6f), 1.0f - 1e-6f);
        float fy = (c0 + 1.0f) * 48.0f - 0.5f;   // unnorm, size 96
        float fx = (c1 + 1.0f) * 48.0f - 0.5f;
        int iy = min(95, max(0, (int)rintf(fy)));  // round-to-nearest-even
        int ix = min(95, max(0, (int)rintf(fx)));
        float oldy = -1.0f + (2.0f * iy + 1.0f) / 96.0f;
        float oldx = -1.0f + (2.0f * ix + 1.0f) / 96.0f;
        float rc0 = (cy - oldy) * 96.0f;
        float rc1 = (cx - oldx) * 96.0f;
        grid[((size_t)b * CIN0 + 2 * t)     * NPIX + p] = rc0;
        grid[((size_t)b * CIN0 + 2 * t + 1) * NPIX + p] = rc1;
        float ar = fabsf(rc0 * rc1) + 1e-9f;
        areas[t] = ar; iys[t] = iy; ixs[t] = ix;
        tot += ar;
    }
    #pragma unroll
    for (int t = 0; t < 4; ++t) {
        const float s = areas[3 - t] / tot;      // diagonal area swap
        const float* f = feat + (size_t)b * NF * PLR + (size_t)iys[t] * WLR + ixs[t];
        float* g = grid + ((size_t)b * CIN0 + 8 + t * NF) * NPIX + p;
        for (int c = 0; c < NF; ++c)
            g[(size_t)c * NPIX] = f[(size_t)c * PLR] * s;
    }
    grid[((size_t)b * CIN0 + 264) * NPIX + p] = cell[b * 2 + 0] * 96.0f;
    grid[((size_t)b * CIN0 + 265) * NPIX + p] = cell[b * 2 + 1] * 96.0f;
    #pragma unroll
    for (int c = 266; c < CIN0; ++c)
        grid[((size_t)b * CIN0 + c) * NPIX + p] = 0.0f;
}

// ---------------- galerkin layernorm (ddof=1, eps on std), in-place ----------
__global__ void k_gln(float* __restrict__ qkv,
                      const float* __restrict__ kw, const float* __restrict__ kb,
                      const float* __restrict__ vw, const float* __restrict__ vb) {
    int p = blockIdx.x * 256 + threadIdx.x;
    int h = blockIdx.y, b = blockIdx.z;
    if (p >= NPIX) return;
    float* base = qkv + ((size_t)b * (3 * WID) + h * 48) * NPIX + p;
    #pragma unroll
    for (int s = 0; s < 2; ++s) {                 // s=0: k, s=1: v
        float* d = base + (size_t)(16 + 16 * s) * NPIX;
        float vals[16], mean = 0.0f;
        #pragma unroll
        for (int c = 0; c < 16; ++c) { vals[c] = d[(size_t)c * NPIX]; mean += vals[c]; }
        mean *= (1.0f / 16.0f);
        float var = 0.0f;
        #pragma unroll
        for (int c = 0; c < 16; ++c) { float t = vals[c] - mean; var += t * t; }
        var *= (1.0f / 15.0f);                    // ddof = 1
        const float inv = 1.0f / (sqrtf(var) + 1e-5f);
        const float* wg = s ? vw : kw;
        const float* bi = s ? vb : kb;
        #pragma unroll
        for (int c = 0; c < 16; ++c)
            d[(size_t)c * NPIX] = wg[h * 16 + c] * ((vals[c] - mean) * inv) + bi[h * 16 + c];
    }
}

// ---------------- kv = k^T v / N  (per batch/head, LDS-staged) ---------------
__global__ void k_kv(const float* __restrict__ qkv, float* __restrict__ kvm) {
    __shared__ float lk[16][129], lv[16][129];
    const int h = blockIdx.x, b = blockIdx.y, tid = threadIdx.x;
    const float* kb = qkv + ((size_t)b * (3 * WID) + h * 48 + 16) * NPIX;
    const float* vb = kb + (size_t)16 * NPIX;
    const int c = tid >> 4, d = tid & 15;
    float acc = 0.0f;
    for (int n0 = 0; n0 < NPIX; n0 += 128) {
        __syncthreads();
        for (int i = tid; i < 16 * 128; i += 256) {
            int ch = i >> 7, nn = i & 127;
            lk[ch][nn] = kb[(size_t)ch * NPIX + n0 + nn];
            lv[ch][nn] = vb[(size_t)ch * NPIX + n0 + nn];
        }
        __syncthreads();
        #pragma unroll 8
        for (int nn = 0; nn < 128; ++nn)
            acc += lk[c][nn] * lv[d][nn];
    }
    kvm[(((size_t)b * NHEAD + h) * 16 + c) * 16 + d] = acc * (1.0f / (float)NPIX);
}

// ---------------- o = q @ kv + x (WMMA, A = kv^T), 2 tiles/wave --------------
__global__ void __launch_bounds__(256)
k_apply_kv(const float* __restrict__ qkv, const float* __restrict__ kvm,
           const float* __restrict__ xres, float* __restrict__ obuf) {
    const int tid = threadIdx.x;
    const int wave = tid >> 5, lane = tid & 31;
    const int half = lane >> 4, ln = lane & 15;
    const int h = blockIdx.y, b = blockIdx.z;
    const int p0 = (blockIdx.x * 8 + wave) * 32;
    const float* kv = kvm + ((size_t)b * NHEAD + h) * 256;          // [c][d]
    const float* q  = qkv + ((size_t)b * (3 * WID) + h * 48) * NPIX + p0 + ln;

    float ax[4], ay[4], b0x[4], b0y[4], b1x[4], b1y[4];
    #pragma unroll
    for (int u = 0; u < 4; ++u) {
        const int ka = 4 * u + half * 2;
        ax[u]  = kv[ka * 16 + ln];          // A[m=d][k=c] = kv[c][d]
        ay[u]  = kv[(ka + 1) * 16 + ln];
        b0x[u] = q[(size_t)ka * NPIX];
        b0y[u] = q[(size_t)(ka + 1) * NPIX];
        b1x[u] = q[(size_t)ka * NPIX + 16];
        b1y[u] = q[(size_t)(ka + 1) * NPIX + 16];
    }
    v8f acc0 = {}, acc1 = {};
    #pragma unroll
    for (int u = 0; u < 4; ++u) {
        v2f a  = {ax[u],  ay[u]};
        v2f f0 = {b0x[u], b0y[u]};
        v2f f1 = {b1x[u], b1y[u]};
        acc0 = wmma4(a, f0, acc0);
        acc1 = wmma4(a, f1, acc1);
    }
    #pragma unroll
    for (int r = 0; r < 8; ++r) {
        const int ch = h * 16 + r + half * 8;
        size_t o = ((size_t)b * WID + ch) * NPIX + p0 + ln;
        obuf[o]      = acc0[r] + xres[o];
        obuf[o + 16] = acc1[r] + xres[o + 16];
    }
}

// ---------------- fc2 (256->3) + bilinear-border residual --------------------
__global__ void k_fc2_out(const float* __restrict__ t1, const float* __restrict__ w2,
                          const float* __restrict__ b2, const float* __restrict__ inp,
                          const float* __restrict__ coord, float* __restrict__ out) {
    __shared__ float wl[3 * 256];
    for (int i = threadIdx.x; i < 3 * 256; i += 256) wl[i] = w2[i];
    __syncthreads();
    int p = blockIdx.x * 256 + threadIdx.x;
    int b = blockIdx.y;
    if (p >= NPIX) return;
    float a0 = b2[0], a1 = b2[1], a2 = b2[2];
    const float* x = t1 + (size_t)b * WID * NPIX + p;
    #pragma unroll 8
    for (int c = 0; c < 256; ++c) {
        float v = x[(size_t)c * NPIX];
        a0 += wl[c] * v; a1 += wl[256 + c] * v; a2 += wl[512 + c] * v;
    }
    const float cy = coord[((size_t)b * NPIX + p) * 2 + 0];
    const float cx = coord[((size_t)b * NPIX + p) * 2 + 1];
    float fy = fminf(fmaxf((cy + 1.0f) * 48.0f - 0.5f, 0.0f), 95.0f);
    float fx = fminf(fmaxf((cx + 1.0f) * 48.0f - 0.5f, 0.0f), 95.0f);
    int y0 = (int)floorf(fy), x0 = (int)floorf(fx);
    float wy = fy - y0, wx = fx - x0;
    int y1 = min(y0 + 1, 95), x1 = min(x0 + 1, 95);
    const float* im = inp + (size_t)b * 3 * PLR;
    float res[3] = {a0, a1, a2};
    #pragma unroll
    for (int c = 0; c < 3; ++c) {
        const float* f = im + (size_t)c * PLR;
        float v00 = f[y0 * WLR + x0], v01 = f[y0 * WLR + x1];
        float v10 = f[y1 * WLR + x0], v11 = f[y1 * WLR + x1];
        float bi = v00 * (1 - wy) * (1 - wx) + v01 * (1 - wy) * wx +
                   v10 * wy * (1 - wx)       + v11 * wy * wx;
        out[((size_t)b * 3 + c) * NPIX + p] = res[c] + bi;
    }
}

// ============================== host orchestration ===========================
extern "C" void kernel_launch(void* const* d_in, const int* in_sizes, int n_in,
                              void* d_out, int out_size, void* d_ws, size_t ws_size,
                              hipStream_t stream) {
    (void)in_sizes; (void)n_in; (void)out_size; (void)ws_size;
    const float* inp      = (const float*)d_in[0];
    const float* coord    = (const float*)d_in[1];
    const float* cell     = (const float*)d_in[2];
    const float* head_w   = (const float*)d_in[3];
    const float* head_b   = (const float*)d_in[4];
    const float* rb_w1    = (const float*)d_in[5];
    const float* rb_b1    = (const float*)d_in[6];
    const float* rb_w2    = (const float*)d_in[7];
    const float* rb_b2    = (const float*)d_in[8];
    const float* tail_w   = (const float*)d_in[9];
    const float* tail_b   = (const float*)d_in[10];
    const float* conv00_w = (const float*)d_in[11];
    const float* conv00_b = (const float*)d_in[12];
    const float* qkv_w    = (const float*)d_in[13];
    const float* qkv_b    = (const float*)d_in[14];
    const float* o1_w     = (const float*)d_in[15];
    const float* o1_b     = (const float*)d_in[16];
    const float* o2_w     = (const float*)d_in[17];
    const float* o2_b     = (const float*)d_in[18];
    const float* kln_w    = (const float*)d_in[19];
    const float* kln_b    = (const float*)d_in[20];
    const float* vln_w    = (const float*)d_in[21];
    const float* vln_b    = (const float*)d_in[22];
    const float* fc1_w    = (const float*)d_in[23];
    const float* fc1_b    = (const float*)d_in[24];
    const float* fc2_w    = (const float*)d_in[25];
    const float* fc2_b    = (const float*)d_in[26];
    float* out = (float*)d_out;

    // workspace carve-up (floats)
    float* ws = (float*)d_ws;
    size_t off = 0;
    auto alloc = [&](size_t n) { float* p = ws + off; off += n; return p; };
    float* padbuf = alloc((size_t)B_ * NF * PADA);     // 3x3-conv staging (also head)
    float* xh     = alloc((size_t)B_ * NF * PLR);
    float* res    = alloc((size_t)B_ * NF * PLR);
    float* tmp    = alloc((size_t)B_ * NF * PLR);
    float* feat   = alloc((size_t)B_ * NF * PLR);
    float* wpad   = alloc((size_t)WID * CIN0);
    float* grid   = alloc((size_t)B_ * CIN0 * NPIX);
    float* xb     = alloc((size_t)B_ * WID * NPIX);
    float* qkv    = alloc((size_t)B_ * 3 * WID * NPIX);
    float* kvm    = alloc((size_t)B_ * NHEAD * 256);
    float* obuf   = alloc((size_t)B_ * WID * NPIX);
    float* t1     = alloc((size_t)B_ * WID * NPIX);

    const dim3 blk(256);
    auto cdiv = [](int a, int b) { return (a + b - 1) / b; };

    // --- encoder ---
    {
        int tot3 = B_ * 3 * (int)PADA;
        k_pad<<<cdiv(tot3, 256), blk, 0, stream>>>(inp, padbuf, 3, tot3);
        k_head<<<cdiv(B_ * NF * PLR, 256), blk, 0, stream>>>(padbuf, head_w, head_b, xh, res);
    }
    const int tot64 = B_ * NF * (int)PADA;
    const dim3 cgrid(36, 4, B_);                // 288 pixel pairs / 8 waves, 4 co tiles
    for (int i = 0; i < 16; ++i) {
        k_pad<<<cdiv(tot64, 256), blk, 0, stream>>>(res, padbuf, NF, tot64);
        k_conv3<0><<<cgrid, blk, 0, stream>>>(padbuf, rb_w1 + (size_t)i * NF * 576,
                                              rb_b1 + (size_t)i * NF, nullptr, tmp);
        k_pad<<<cdiv(tot64, 256), blk, 0, stream>>>(tmp, padbuf, NF, tot64);
        k_conv3<1><<<cgrid, blk, 0, stream>>>(padbuf, rb_w2 + (size_t)i * NF * 576,
                                              rb_b2 + (size_t)i * NF, res, res);
    }
    k_pad<<<cdiv(tot64, 256), blk, 0, stream>>>(res, padbuf, NF, tot64);
    k_conv3<1><<<cgrid, blk, 0, stream>>>(padbuf, tail_w, tail_b, xh, feat);

    // --- grid assembly + conv00 ---
    k_make_grid<<<dim3(NPIX / 256, B_), blk, 0, stream>>>(coord, cell, feat, grid);
    k_padw<<<cdiv(WID * CIN0, 256), blk, 0, stream>>>(conv00_w, wpad);
    k_gemm<0, false><<<dim3(NPIX / 256, WID / 16, B_), blk, 0, stream>>>(
        wpad, conv00_b, grid, nullptr, xb, CIN0, WID);

    // --- two galerkin attention blocks ---
    for (int l = 0; l < 2; ++l) {
        const float* qw = qkv_w + (size_t)l * 3 * WID * WID;
        const float* qb = qkv_b + (size_t)l * 3 * WID;
        k_gemm<0, false><<<dim3(NPIX / 256, (3 * WID) / 16, B_), blk, 0, stream>>>(
            qw, qb, xb, nullptr, qkv, WID, 3 * WID);
        k_gln<<<dim3(NPIX / 256, NHEAD, B_), blk, 0, stream>>>(
            qkv, kln_w + l * 256, kln_b + l * 256, vln_w + l * 256, vln_b + l * 256);
        k_kv<<<dim3(NHEAD, B_), blk, 0, stream>>>(qkv, kvm);
        k_apply_kv<<<dim3(NPIX / 256, NHEAD, B_), blk, 0, stream>>>(qkv, kvm, xb, obuf);
        k_gemm<1, false><<<dim3(NPIX / 256, WID / 16, B_), blk, 0, stream>>>(
            o1_w + (size_t)l * WID * WID, o1_b + (size_t)l * WID, obuf, nullptr, t1,
            WID, WID);
        k_gemm<0, true><<<dim3(NPIX / 256, WID / 16, B_), blk, 0, stream>>>(
            o2_w + (size_t)l * WID * WID, o2_b + (size_t)l * WID, t1, obuf, xb,
            WID, WID);
    }

    // --- output MLP + bilinear residual ---
    k_gemm<1, false><<<dim3(NPIX / 256, WID / 16, B_), blk, 0, stream>>>(
        fc1_w, fc1_b, xb, nullptr, t1, WID, WID);
    k_fc2_out<<<dim3(NPIX / 256, B_), blk, 0, stream>>>(t1, fc2_w, fc2_b, inp, coord, out);
}